// Engel2022Fit_68925635166721
// MI455X (gfx1250) — compile-verified
//
#include <hip/hip_runtime.h>

typedef float v2f __attribute__((ext_vector_type(2)));
typedef float v8f __attribute__((ext_vector_type(8)));

#define B_  256
#define T_  2048
#define BT  (B_ * T_)   // 524288 rows of the flattened [B*T] hidden/output

// ---------------------------------------------------------------------------
// Kernel 1: Cayley parametrization, but only columns 0,1 of Q.
// Q[:,k] = (I - A) v_k with (I + A) v_k = e_k,  A = matB - matB^T (skew).
// Symmetric part of (I+A) is I (SPD) -> Gauss-Jordan without pivoting is stable.
// One workgroup of 128 threads (one row per thread), augmented matrix
// [128 x 130] in dynamic LDS with stride 131 (131 % 64 = 3 -> <=2-way conflicts).
// ---------------------------------------------------------------------------
__global__ void cayley_q2_kernel(const float* __restrict__ matB,
                                 float2* __restrict__ q2) {
  extern __shared__ float lds[];           // [128][131] + vbuf[256]
  const int S = 131;
  const int j = threadIdx.x;               // row 0..127
  float* vbuf = lds + 128 * S;

  // Build row j of M = I + A, augmented with e0, e1 in cols 128,129.
  for (int c = 0; c < 128; ++c) {
    float a = matB[j * 128 + c] - matB[c * 128 + j];
    lds[j * S + c] = a + (c == j ? 1.0f : 0.0f);
  }
  lds[j * S + 128] = (j == 0) ? 1.0f : 0.0f;
  lds[j * S + 129] = (j == 1) ? 1.0f : 0.0f;
  __syncthreads();

  // Gauss-Jordan: eliminate column k from all rows j != k.
  for (int k = 0; k < 128; ++k) {
    float piv = lds[k * S + k];
    if (j != k) {
      float f = lds[j * S + k] / piv;
      for (int c = k + 1; c < 130; ++c)
        lds[j * S + c] -= f * lds[k * S + c];
    }
    __syncthreads();
  }

  float dj = lds[j * S + j];
  float v0 = lds[j * S + 128] / dj;        // inv(I+A) e0, row j
  float v1 = lds[j * S + 129] / dj;        // inv(I+A) e1, row j
  vbuf[j] = v0;
  vbuf[128 + j] = v1;
  __syncthreads();

  // q_k[j] = v_k[j] - (A v_k)[j]
  float q0 = v0, q1 = v1;
  for (int i = 0; i < 128; ++i) {
    float a = matB[j * 128 + i] - matB[i * 128 + j];
    q0 -= a * vbuf[i];
    q1 -= a * vbuf[128 + i];
  }
  q2[j] = make_float2(q0, q1);
}

// ---------------------------------------------------------------------------
// Kernel 2: pre-scaled input projection xw[r] = ALPHA * (u[r] @ wIn^T).
// Fully parallel over B*T; folding ALPHA here shortens the serial chain in k3.
// ---------------------------------------------------------------------------
__global__ void input_proj_kernel(const float* __restrict__ u,
                                  const float* __restrict__ wIn,
                                  float2* __restrict__ xw) {
  int r = blockIdx.x * blockDim.x + threadIdx.x;
  if (r >= BT) return;
  float u0 = u[r * 3 + 0], u1 = u[r * 3 + 1], u2 = u[r * 3 + 2];
  float x0 = 0.1f * (u0 * wIn[0] + u1 * wIn[1] + u2 * wIn[2]);
  float x1 = 0.1f * (u0 * wIn[3] + u1 * wIn[4] + u2 * wIn[5]);
  xw[r] = make_float2(x0, x1);
}

// ---------------------------------------------------------------------------
// Kernel 3: the sequential leaky-RNN scan. One batch per lane, 8 single-wave
// workgroups (spread across WGPs). In-place: reads xw[b][t], writes y[b][t]
// back to the same slot. Dependent chain per step: max -> fma -> add (~3 deep).
//   y' = 0.9*y + (0.1*wRec) @ relu(y) + xw    (xw already has the 0.1 factor)
// ---------------------------------------------------------------------------
__global__ void rnn_scan_kernel(const float* __restrict__ wRec,
                                float4* __restrict__ xwh) {
  int b = blockIdx.x * 32 + threadIdx.x;   // 8 blocks x 32 lanes = 256 batches
  float w00 = 0.1f * wRec[0], w01 = 0.1f * wRec[1];
  float w10 = 0.1f * wRec[2], w11 = 0.1f * wRec[3];
  float y0 = 0.0f, y1 = 0.0f;
  float4* p = xwh + (size_t)b * (T_ / 2);  // 2 timesteps per float4
  for (int i = 0; i < T_ / 2; ++i) {
    float4 v = p[i];
    // step t
    float r0 = fmaxf(y0, 0.0f), r1 = fmaxf(y1, 0.0f);
    float n0 = fmaf(w00, r0, 0.9f * y0) + fmaf(w01, r1, v.x);
    float n1 = fmaf(w10, r0, 0.9f * y1) + fmaf(w11, r1, v.y);
    float o0 = n0, o1 = n1;
    // step t+1
    r0 = fmaxf(n0, 0.0f); r1 = fmaxf(n1, 0.0f);
    y0 = fmaf(w00, r0, 0.9f * n0) + fmaf(w01, r1, v.z);
    y1 = fmaf(w10, r0, 0.9f * n1) + fmaf(w11, r1, v.w);
    p[i] = make_float4(o0, o1, y0, y1);    // hidden states overwrite xw in place
  }
}

// ---------------------------------------------------------------------------
// Kernel 4: rank-2 expansion out[r][j] = h0[r]*Q[j][0] + h1[r]*Q[j][1]
// via V_WMMA_F32_16X16X4_F32. One wave per 16-row block x 128 cols (8 tiles).
// A (16x4 f32): VGPR0 = K0 rows (lanes 0-15 = M), VGPR1 = K1; K2/K3 slots
// (lanes 16-31) zero.  B (4x16): VGPR0 = K0 row (lanes 0-15 = N), VGPR1 = K1;
// lanes 16-31 (K2/K3) zero.  D layout: VGPR i -> row M=i (lanes 0-15) /
// M=i+8 (lanes 16-31), col N = lane&15.
// ---------------------------------------------------------------------------
__global__ void expand_wmma_kernel(const float2* __restrict__ hid,
                                   const float2* __restrict__ q2,
                                   float* __restrict__ out) {
  int lane   = threadIdx.x & 31;
  int wid    = threadIdx.x >> 5;
  int rb     = blockIdx.x * 8 + wid;       // 16-row block index
  int base   = rb * 16;                    // flattened row = b*T + t
  int lane16 = lane & 15;
  int hi     = lane >> 4;

  float a0 = 0.0f, a1 = 0.0f;
  if (lane < 16) {
    float2 h = hid[base + lane];
    a0 = h.x; a1 = h.y;
  }
  v2f A;
  A[0] = a0;   // K = 0 component (h0)
  A[1] = a1;   // K = 1 component (h1)

  float bq0[8], bq1[8];
#pragma unroll
  for (int jb = 0; jb < 8; ++jb) {
    float b0 = 0.0f, b1 = 0.0f;
    if (lane < 16) {
      float2 q = q2[jb * 16 + lane];       // (Q[j][0], Q[j][1])
      b0 = q.x; b1 = q.y;
    }
    bq0[jb] = b0; bq1[jb] = b1;
  }

  float* orow = out + (size_t)(base + hi * 8) * 128 + lane16;
#pragma unroll
  for (int jb = 0; jb < 8; ++jb) {
    v2f Bv;
    Bv[0] = bq0[jb];
    Bv[1] = bq1[jb];
    v8f c = {};
    v8f d = __builtin_amdgcn_wmma_f32_16x16x4_f32(
        /*neg_a=*/false, A, /*neg_b=*/false, Bv,
        /*c_mod=*/(short)0, c, /*reuse_a=*/false, /*reuse_b=*/false);
    float* o = orow + jb * 16;
#pragma unroll
    for (int i = 0; i < 8; ++i)
      o[(size_t)i * 128] = d[i];
  }
}

// ---------------------------------------------------------------------------
// Launch: Q-solve -> input projection -> serial scan -> WMMA expansion.
// Workspace: [0, 4MiB)   = xw / hidden (in place, float2 per (b,t))
//            [4MiB, +1KB) = q2 (128 x float2)
// ---------------------------------------------------------------------------
extern "C" void kernel_launch(void* const* d_in, const int* in_sizes, int n_in,
                              void* d_out, int out_size, void* d_ws, size_t ws_size,
                              hipStream_t stream) {
  const float* u    = (const float*)d_in[0];   // [256,2048,3]
  const float* matB = (const float*)d_in[1];   // [128,128]
  const float* wIn  = (const float*)d_in[2];   // [2,3]
  const float* wRec = (const float*)d_in[3];   // [2,2]
  float* out = (float*)d_out;                  // [256,2048,128]

  float*  wsf = (float*)d_ws;
  float2* xwh = (float2*)wsf;                  // BT float2 = 4 MiB
  float2* q2  = (float2*)(wsf + 2 * BT);       // 128 float2

  size_t ldsBytes = (size_t)(128 * 131 + 256) * sizeof(float);  // ~68 KB (<320KB WGP LDS)
  cayley_q2_kernel<<<1, 128, ldsBytes, stream>>>(matB, q2);
  input_proj_kernel<<<BT / 256, 256, 0, stream>>>(u, wIn, xwh);
  rnn_scan_kernel<<<8, 32, 0, stream>>>(wRec, (float4*)xwh);
  expand_wmma_kernel<<<BT / 16 / 8, 256, 0, stream>>>(xwh, q2, out);
}